// CrfRnnLayer_1030792151717
// MI455X (gfx1250) — compile-verified
//
#include <hip/hip_runtime.h>
#include <hip/hip_bf16.h>

typedef __attribute__((ext_vector_type(2))) float v2f;
typedef __attribute__((ext_vector_type(8))) float v8f;

#define HH 32
#define WW 16
#define DD 16
#define CC 4
#define NN (HH*WW*DD)      // 8192
#define NT (NN/16)         // 512 tiles of 16 rows
#define TH_GAMMA 3.0f
#define TH_ALPHA 8.0f
#define TH_BETA  0.5f
#define LOG2E 1.44269504088896340736f

// ---------------------------------------------------------------------------
// Precompute augmented features in WMMA A/B VGPR layout.
// K=8 augmented dims. A-side aug: LOG2E * [f..., 1, -0.5*sq]; B-side:
// [f..., -0.5*sq, 1], so that A_i . B_j = log2(e) * (f_i.f_j - 0.5|f_i|^2
// - 0.5|f_j|^2) and exp2() gives the Gaussian kernel directly.
// Memory layout: [tile][lane][slot] so each lane loads its 4 slots as b128.
// Slot s, lane l (m = l&15, hi = l>>4) holds k = (s&1) + 2*hi + 4*(s>>1),
// matching the V_WMMA_F32_16X16X4_F32 operand layout (2 VGPRs per K-quad).
// ---------------------------------------------------------------------------
__global__ void crf_precompute_feats(const float* __restrict__ rgb,
                                     float* __restrict__ Asp, float* __restrict__ Bsp,
                                     float* __restrict__ Abi, float* __restrict__ Bbi) {
    int tile = blockIdx.x;
    int t = threadIdx.x;          // 0..127
    int s = t >> 5;               // slot
    int l = t & 31;               // lane
    int m = l & 15;
    int hi = l >> 4;
    int k = (s & 1) + 2 * hi + 4 * (s >> 1);   // 0..7
    int n = tile * 16 + m;
    int d  = n & 15;
    int w  = (n >> 4) & 15;
    int h  = n >> 8;
    float px = (float)(h + 1), py = (float)(w + 1), pz = (float)(d + 1);

    // spatial features (dim 3)
    float f0 = px / TH_GAMMA, f1 = py / TH_GAMMA, f2 = pz / TH_GAMMA;
    float sqS = f0 * f0 + f1 * f1 + f2 * f2;
    // bilateral features (dim 6)
    float g0 = px / TH_ALPHA, g1 = py / TH_ALPHA, g2 = pz / TH_ALPHA;
    float r0 = rgb[n * 3 + 0] / TH_BETA;
    float r1 = rgb[n * 3 + 1] / TH_BETA;
    float r2 = rgb[n * 3 + 2] / TH_BETA;
    float sqB = g0 * g0 + g1 * g1 + g2 * g2 + r0 * r0 + r1 * r1 + r2 * r2;

    float fs;
    fs = (k == 0) ? f0 : (k == 1) ? f1 : (k == 2) ? f2 : 0.0f;
    float aS = (k < 3) ? fs : (k == 3 ? 1.0f : (k == 4 ? -0.5f * sqS : 0.0f));
    float bS = (k < 3) ? fs : (k == 3 ? -0.5f * sqS : (k == 4 ? 1.0f : 0.0f));

    float gs;
    gs = (k == 0) ? g0 : (k == 1) ? g1 : (k == 2) ? g2 :
         (k == 3) ? r0 : (k == 4) ? r1 : (k == 5) ? r2 : 0.0f;
    float aB = (k < 6) ? gs : (k == 6 ? 1.0f : -0.5f * sqB);
    float bB = (k < 6) ? gs : (k == 6 ? -0.5f * sqB : 1.0f);

    int idx = (tile * 32 + l) * 4 + s;   // [tile][lane][slot]
    Asp[idx] = aS * LOG2E; Bsp[idx] = bS;
    Abi[idx] = aB * LOG2E; Bbi[idx] = bB;
}

// ---------------------------------------------------------------------------
// Row softmax over C=4 classes.
// ---------------------------------------------------------------------------
__global__ void crf_softmax_rows(const float* __restrict__ q, float* __restrict__ sm) {
    int n = blockIdx.x * blockDim.x + threadIdx.x;
    if (n >= NN) return;
    float4 v = *(const float4*)(q + n * 4);
    float mx = fmaxf(fmaxf(v.x, v.y), fmaxf(v.z, v.w));
    float e0 = __expf(v.x - mx), e1 = __expf(v.y - mx);
    float e2 = __expf(v.z - mx), e3 = __expf(v.w - mx);
    float inv = 1.0f / (e0 + e1 + e2 + e3);
    float4 o; o.x = e0 * inv; o.y = e1 * inv; o.z = e2 * inv; o.w = e3 * inv;
    *(float4*)(sm + n * 4) = o;
}

// ---------------------------------------------------------------------------
// Fused dual attention + message passing + compatibility + unary add.
// One block (4 waves) per 16-row i-tile; each wave handles every 4th j-tile.
// Logits via V_WMMA_F32_16X16X4_F32 (two per attention, K=8 augmented feats,
// log2-domain); exp2 + P@sm accumulated per-lane directly in WMMA D layout:
//   D: vgpr v, lane l -> row i = v + 8*(l>=16), col j = l&15.
// Final 16-lane x 4-wave reduction through LDS.
// ---------------------------------------------------------------------------
__global__ __launch_bounds__(128) void crf_attention_step(
    const float* __restrict__ Asp, const float* __restrict__ Bsp,
    const float* __restrict__ Abi, const float* __restrict__ Bbi,
    const float* __restrict__ sm, const float* __restrict__ u,
    const float* __restrict__ SK, const float* __restrict__ BK,
    const float* __restrict__ CM, float* __restrict__ qout) {

    int tile = blockIdx.x;
    int lane = threadIdx.x & 31;
    int wv   = threadIdx.x >> 5;

    // A operands for this i-tile (identical for all 4 waves), one b128 each
    float4 avS = *(const float4*)(Asp + (tile * 32 + lane) * 4);
    float4 avB = *(const float4*)(Abi + (tile * 32 + lane) * 4);
    v2f aS0, aS1, aB0, aB1;
    aS0[0] = avS.x; aS0[1] = avS.y; aS1[0] = avS.z; aS1[1] = avS.w;
    aB0[0] = avB.x; aB0[1] = avB.y; aB1[0] = avB.z; aB1[1] = avB.w;

    float numS[8][4] = {}, numB[8][4] = {};
    float denS[8] = {}, denB[8] = {};

    for (int tj = wv; tj < NT; tj += 4) {
        float4 bvS = *(const float4*)(Bsp + (tj * 32 + lane) * 4);
        float4 bvB = *(const float4*)(Bbi + (tj * 32 + lane) * 4);
        v2f bS0, bS1, bB0, bB1;
        bS0[0] = bvS.x; bS0[1] = bvS.y; bS1[0] = bvS.z; bS1[1] = bvS.w;
        bB0[0] = bvB.x; bB0[1] = bvB.y; bB1[0] = bvB.z; bB1[1] = bvB.w;

        v8f cS = {};
        cS = __builtin_amdgcn_wmma_f32_16x16x4_f32(false, aS0, false, bS0, (short)0, cS, false, false);
        cS = __builtin_amdgcn_wmma_f32_16x16x4_f32(false, aS1, false, bS1, (short)0, cS, false, false);
        v8f cB = {};
        cB = __builtin_amdgcn_wmma_f32_16x16x4_f32(false, aB0, false, bB0, (short)0, cB, false, false);
        cB = __builtin_amdgcn_wmma_f32_16x16x4_f32(false, aB1, false, bB1, (short)0, cB, false, false);

        int j = tj * 16 + (lane & 15);
        float4 smj = *(const float4*)(sm + j * 4);

#pragma unroll
        for (int v = 0; v < 8; ++v) {
            float pS = __builtin_amdgcn_exp2f(cS[v]);  // log2-domain logits <= 0
            denS[v]    += pS;
            numS[v][0] += pS * smj.x;
            numS[v][1] += pS * smj.y;
            numS[v][2] += pS * smj.z;
            numS[v][3] += pS * smj.w;
            float pB = __builtin_amdgcn_exp2f(cB[v]);
            denB[v]    += pB;
            numB[v][0] += pB * smj.x;
            numB[v][1] += pB * smj.y;
            numB[v][2] += pB * smj.z;
            numB[v][3] += pB * smj.w;
        }
    }

    // ---- reduction across 16 lanes x 4 waves ----
    __shared__ float part[4][32][80];
    __shared__ float souts[16][4], bouts[16][4];
    float* pp = part[wv][lane];
#pragma unroll
    for (int v = 0; v < 8; ++v) {
        pp[v * 4 + 0]      = numS[v][0];
        pp[v * 4 + 1]      = numS[v][1];
        pp[v * 4 + 2]      = numS[v][2];
        pp[v * 4 + 3]      = numS[v][3];
        pp[32 + v]         = denS[v];
        pp[40 + v * 4 + 0] = numB[v][0];
        pp[40 + v * 4 + 1] = numB[v][1];
        pp[40 + v * 4 + 2] = numB[v][2];
        pp[40 + v * 4 + 3] = numB[v][3];
        pp[72 + v]         = denB[v];
    }
    __syncthreads();

    int t = threadIdx.x;
    if (t < 64) {
        int r = t >> 2, c = t & 3;
        int hb = (r >= 8) ? 16 : 0;   // rows 8..15 live in lanes 16..31
        int v = r & 7;
        float nS = 0.f, dS = 0.f, nB = 0.f, dB = 0.f;
        for (int w = 0; w < 4; ++w) {
            for (int dl = 0; dl < 16; ++dl) {
                const float* p = part[w][hb + dl];
                nS += p[v * 4 + c];
                dS += p[32 + v];
                nB += p[40 + v * 4 + c];
                dB += p[72 + v];
            }
        }
        souts[r][c] = nS / dS;
        bouts[r][c] = nB / dB;
    }
    __syncthreads();

    if (t < 64) {
        int r = t >> 2, c = t & 3;
        float pw = 0.f;
#pragma unroll
        for (int d2 = 0; d2 < 4; ++d2) {
            float msg = 0.f;
#pragma unroll
            for (int e = 0; e < 4; ++e)
                msg += SK[d2 * 4 + e] * souts[r][e] + BK[d2 * 4 + e] * bouts[r][e];
            pw += CM[c * 4 + d2] * msg;
        }
        int n = tile * 16 + r;
        qout[n * 4 + c] = u[n * 4 + c] + pw;
    }
}

// ---------------------------------------------------------------------------
extern "C" void kernel_launch(void* const* d_in, const int* in_sizes, int n_in,
                              void* d_out, int out_size, void* d_ws, size_t ws_size,
                              hipStream_t stream) {
    const float* u   = (const float*)d_in[0];   // unaries [N,4]
    const float* rgb = (const float*)d_in[1];   // [N,3]
    const float* SK  = (const float*)d_in[2];   // [4,4]
    const float* BK  = (const float*)d_in[3];   // [4,4]
    const float* CM  = (const float*)d_in[4];   // [4,4]
    float* out = (float*)d_out;

    float* ws = (float*)d_ws;
    float* sm  = ws;                 // N*4
    float* q   = sm  + NN * 4;       // N*4
    float* Asp = q   + NN * 4;       // NT*32*4
    float* Bsp = Asp + NT * 32 * 4;
    float* Abi = Bsp + NT * 32 * 4;
    float* Bbi = Abi + NT * 32 * 4;

    crf_precompute_feats<<<NT, 128, 0, stream>>>(rgb, Asp, Bsp, Abi, Bbi);

    // iteration 0: softmax(u) -> attention -> q
    crf_softmax_rows<<<(NN + 255) / 256, 256, 0, stream>>>(u, sm);
    crf_attention_step<<<NT, 128, 0, stream>>>(Asp, Bsp, Abi, Bbi, sm, u, SK, BK, CM, q);
    // iteration 1: softmax(q) -> attention -> out
    crf_softmax_rows<<<(NN + 255) / 256, 256, 0, stream>>>(q, sm);
    crf_attention_step<<<NT, 128, 0, stream>>>(Asp, Bsp, Abi, Bbi, sm, u, SK, BK, CM, out);
}